// GCN_55997783605387
// MI455X (gfx1250) — compile-verified
//
#include <hip/hip_runtime.h>
#include <hip/hip_bf16.h>

#define D 128
#define MTILES 4              // 16-row m-tiles per GEMM block (64 nodes/block)
#define ASTRIDE 132           // padded LDS row stride (floats): 132%64=4 -> conflict-free A reads

typedef __attribute__((ext_vector_type(2))) float v2f;
typedef __attribute__((ext_vector_type(8))) float v8f;

// ---------------------------------------------------------------------------
// CDNA5 async global->LDS copy (bypasses VGPRs, tracked by ASYNCcnt).
// Per-lane: copies 16 bytes from the global address in gptr to the LDS address in lds_off.
// ---------------------------------------------------------------------------
__device__ __forceinline__ void async_load_b128(unsigned lds_off, const float* gptr) {
  asm volatile("global_load_async_to_lds_b128 %0, %1, off"
               :: "v"(lds_off), "v"(gptr)
               : "memory");
}
__device__ __forceinline__ void wait_asynccnt0() {
  asm volatile("s_wait_asynccnt 0x0" ::: "memory");
}

// ---------------------------------------------------------------------------
// Degree computation: one thread per edge, native f32 atomics into L2.
// ---------------------------------------------------------------------------
__global__ void gcn_degree_kernel(const int* __restrict__ src,
                                  const int* __restrict__ dst,
                                  float* __restrict__ deg_src,
                                  float* __restrict__ deg_dst, int E) {
  int e = blockIdx.x * blockDim.x + threadIdx.x;
  if (e < E) {
    unsafeAtomicAdd(&deg_src[src[e]], 1.0f);
    unsafeAtomicAdd(&deg_dst[dst[e]], 1.0f);
  }
}

// ---------------------------------------------------------------------------
// SpMM: 32 threads per edge, float4 per thread (32*16B = 512B = one D=128 row).
// Gather h[src]*rsqrt(out_deg[src]) and scatter-add into agg[dst] with
// global_atomic_add_f32 (L2-resident: 51MB feature matrix << 192MB L2).
// ---------------------------------------------------------------------------
__global__ void gcn_spmm_kernel(const float* __restrict__ h,
                                const int* __restrict__ src,
                                const int* __restrict__ dst,
                                const float* __restrict__ deg_src,
                                float* __restrict__ agg, int E) {
  long long t = (long long)blockIdx.x * blockDim.x + threadIdx.x;
  long long e = t >> 5;          // edge index
  int c = (int)(t & 31);         // float4 chunk within the 128-wide row
  if (e >= E) return;
  int s = src[e];
  int d = dst[e];
  float ns = rsqrtf(fmaxf(deg_src[s], 1.0f));
  const float4* hp = (const float4*)(h + (long long)s * D);
  float4 v = hp[c];
  float* ap = agg + (long long)d * D + c * 4;
  unsafeAtomicAdd(ap + 0, v.x * ns);
  unsafeAtomicAdd(ap + 1, v.y * ns);
  unsafeAtomicAdd(ap + 2, v.z * ns);
  unsafeAtomicAdd(ap + 3, v.w * ns);
}

// ---------------------------------------------------------------------------
// Dense stage: out = relu( (agg * rsqrt(in_deg)) @ W + b )
//
// Block = 8 waves, each wave owns a 16-column slice of the output.
//  * B fragments (wave's 16 cols, full K=128) preloaded ONCE into 64 VGPRs,
//    reused across MTILES row tiles  -> W L2 traffic / MTILES.
//  * Each 16x128 agg row-tile staged ONCE per block into LDS via
//    global_load_async_to_lds_b128 -> agg read 1x instead of 8x.
//  * 32 x V_WMMA_F32_16X16X4_F32 per tile; norm_dst folded into A fragment,
//    bias+relu folded into D store.
//
// Fragment layouts per CDNA5 ISA 7.12.2 (f32):
//   A 16x4 : lane L<16 -> M=L, K={0,1} in v0,v1 ; lane L>=16 -> M=L-16, K={2,3}
//   B 4x16 : lane L<16 -> N=L, K={0,1}          ; lane L>=16 -> N=L-16, K={2,3}
//   C/D    : VGPR r: lanes 0-15 -> (M=r, N=lane); lanes 16-31 -> (M=r+8, N=lane-16)
// ---------------------------------------------------------------------------
__global__ __launch_bounds__(256) void gcn_gemm_relu_wmma(
    const float* __restrict__ agg, const float* __restrict__ deg_dst,
    const float* __restrict__ W, const float* __restrict__ bias,
    float* __restrict__ out, int N) {
  __shared__ float atile[16 * ASTRIDE];  // padded 16x128 tile, 8448 B

  const int lane = threadIdx.x & 31;
  const int wv   = threadIdx.x >> 5;   // 0..7 -> output-column tile
  const int n0   = wv << 4;
  const int mr   = lane & 15;
  const int kh   = (lane >> 4) << 1;   // 0 for lanes 0-15, 2 for lanes 16-31
  const int n    = n0 + mr;

  // ---- preload this wave's B slice (full K) into registers -----------------
  v2f bf[32];
#pragma unroll
  for (int kk = 0; kk < 32; ++kk) {
    const int k = kk << 2;
    bf[kk].x = W[(k + kh)     * D + n];
    bf[kk].y = W[(k + kh + 1) * D + n];
  }
  const float bb = bias[n];

  // staging assignment: 512 float4 per tile, 2 per thread
  const int idx0 = threadIdx.x;        // float4 index 0..255
  const int idx1 = threadIdx.x + 256;  // float4 index 256..511

  for (int t = 0; t < MTILES; ++t) {
    const int m0 = (blockIdx.x * MTILES + t) << 4;

    // ---- async stage agg[m0..m0+15][0..127] into LDS -----------------------
    {
      int r0 = idx0 >> 5, c0 = idx0 & 31;           // row, float4-col
      int r1 = idx1 >> 5, c1 = idx1 & 31;
      long long gr0 = m0 + r0; if (gr0 >= N) gr0 = N - 1;   // clamp (safe re-read)
      long long gr1 = m0 + r1; if (gr1 >= N) gr1 = N - 1;
      unsigned lbase = (unsigned)(size_t)atile;
      async_load_b128(lbase + (unsigned)(r0 * ASTRIDE + c0 * 4) * 4,
                      agg + gr0 * D + c0 * 4);
      async_load_b128(lbase + (unsigned)(r1 * ASTRIDE + c1 * 4) * 4,
                      agg + gr1 * D + c1 * 4);
    }
    wait_asynccnt0();
    __syncthreads();

    // ---- compute 16x16 tile ------------------------------------------------
    long long m = m0 + mr; if (m >= N) m = N - 1;  // EXEC stays all-ones
    const float nd = rsqrtf(fmaxf(deg_dst[m], 1.0f));
    const float* __restrict__ arow = atile + mr * ASTRIDE;

    v8f acc = {};
#pragma unroll
    for (int kk = 0; kk < 32; ++kk) {
      const int k = kk << 2;
      v2f a;
      a.x = arow[k + kh]     * nd;
      a.y = arow[k + kh + 1] * nd;
      acc = __builtin_amdgcn_wmma_f32_16x16x4_f32(
          /*neg_a=*/false, a, /*neg_b=*/false, bf[kk],
          /*c_mod=*/(short)0, acc, /*reuse_a=*/false, /*reuse_b=*/false);
    }

    const int mbase = m0 + ((lane >> 4) << 3);
#pragma unroll
    for (int r = 0; r < 8; ++r) {
      int mo = mbase + r;
      if (mo < N) out[(long long)mo * D + n] = fmaxf(acc[r] + bb, 0.0f);
    }
    __syncthreads();  // all reads of atile done before next tile's async write
  }
}

// ---------------------------------------------------------------------------
// Orchestration. Workspace layout:
//   agg     : N*D f32  (51.2 MB)
//   h1      : N*D f32  (51.2 MB)
//   deg_src : N   f32
//   deg_dst : N   f32
// ---------------------------------------------------------------------------
extern "C" void kernel_launch(void* const* d_in, const int* in_sizes, int n_in,
                              void* d_out, int out_size, void* d_ws, size_t ws_size,
                              hipStream_t stream) {
  const float* features = (const float*)d_in[0];
  const int*   src      = (const int*)d_in[1];
  const int*   dst      = (const int*)d_in[2];
  const float* W1       = (const float*)d_in[3];
  const float* b1       = (const float*)d_in[4];
  const float* W2       = (const float*)d_in[5];
  const float* b2       = (const float*)d_in[6];
  float*       out      = (float*)d_out;

  const int N = in_sizes[0] / D;
  const int E = in_sizes[1];

  char* ws = (char*)d_ws;
  float* agg     = (float*)ws;  ws += (size_t)N * D * sizeof(float);
  float* h1      = (float*)ws;  ws += (size_t)N * D * sizeof(float);
  float* deg_src = (float*)ws;  ws += (size_t)N * sizeof(float);
  float* deg_dst = (float*)ws;

  hipMemsetAsync(agg,     0, (size_t)N * D * sizeof(float), stream);
  hipMemsetAsync(deg_src, 0, (size_t)N * sizeof(float), stream);
  hipMemsetAsync(deg_dst, 0, (size_t)N * sizeof(float), stream);

  const dim3 blk(256);
  const int deg_blocks  = (E + 255) / 256;
  const int spmm_blocks = (int)(((long long)E * 32 + 255) / 256);
  const int gemm_blocks = (N + (16 * MTILES) - 1) / (16 * MTILES);

  gcn_degree_kernel<<<deg_blocks, blk, 0, stream>>>(src, dst, deg_src, deg_dst, E);

  // Layer 1
  gcn_spmm_kernel<<<spmm_blocks, blk, 0, stream>>>(features, src, dst, deg_src, agg, E);
  gcn_gemm_relu_wmma<<<gemm_blocks, blk, 0, stream>>>(agg, deg_dst, W1, b1, h1, N);

  // Layer 2 (degrees are graph-only, reuse them)
  hipMemsetAsync(agg, 0, (size_t)N * D * sizeof(float), stream);
  gcn_spmm_kernel<<<spmm_blocks, blk, 0, stream>>>(h1, src, dst, deg_src, agg, E);
  gcn_gemm_relu_wmma<<<gemm_blocks, blk, 0, stream>>>(agg, deg_dst, W2, b2, out, N);
}